// MyLoss_72258529788243
// MI455X (gfx1250) — compile-verified
//
#include <hip/hip_runtime.h>

typedef float v2f __attribute__((ext_vector_type(2)));
typedef float v4f __attribute__((ext_vector_type(4)));
typedef float v8f __attribute__((ext_vector_type(8)));
typedef int   v4i __attribute__((ext_vector_type(4)));

#define WG   256
#define GRID 1000

// Kernel 1: stream the bulk (exactly cnt float4s per thread, no tail -> EXEC
// always all-1s so WMMA is legal). Per iteration each lane folds its 4
// elements into (l1, l0) and accumulates them via V_WMMA_F32_16X16X4_F32:
//   A[m][k]: lane m holds {l1, l0} in K=0,1 ; lane m+16 holds K=2,3
//   B[k][n]: constant one-hot, even K -> column 0, odd K -> column 1
// so D[m][0] accumulates l1 of lanes {m, m+16}, D[m][1] accumulates l0.
__global__ __launch_bounds__(WG) void bce_partial_kernel(
    const float* __restrict__ x, const int* __restrict__ t,
    float* __restrict__ ws, int cnt)
{
    const int tid  = threadIdx.x;
    const int lane = tid & 31;
    const int wave = tid >> 5;
    const long long T = (long long)GRID * WG;
    const long long g = (long long)blockIdx.x * WG + tid;

    // Constant routing matrix B (4x16 f32, 2 VGPRs):
    // VGPR0 holds K=0 (lanes 0-15) / K=2 (lanes 16-31); VGPR1 holds K=1 / K=3.
    v2f bB;
    bB.x = (lane == 0 || lane == 16) ? 1.0f : 0.0f;  // even K -> col 0
    bB.y = (lane == 1 || lane == 17) ? 1.0f : 0.0f;  // odd  K -> col 1

    v8f acc = {0.f, 0.f, 0.f, 0.f, 0.f, 0.f, 0.f, 0.f};
    float tc = 0.0f;

    const v4f* xv = (const v4f*)x;
    const v4i* tv = (const v4i*)t;

    for (int i = 0; i < cnt; ++i) {
        long long idx = g + (long long)i * T;
        v4f xx = __builtin_nontemporal_load(&xv[idx]);   // global_load_b128, TH=NT
        v4i tt = __builtin_nontemporal_load(&tv[idx]);
        float l1 = 0.0f, l0 = 0.0f;
#pragma unroll
        for (int j = 0; j < 4; ++j) {
            float xe = xx[j];
            int   te = tt[j];
            float arg = te ? xe : (1.0f - xe);
            float lg  = __logf(arg);                     // one v_log_f32 per element
            l1 += te ? lg : 0.0f;
            l0 += te ? 0.0f : lg;
            tc += (float)te;
        }
        v2f aA;
        aA.x = l1;
        aA.y = l0;
        acc = __builtin_amdgcn_wmma_f32_16x16x4_f32(
            /*neg_a=*/false, aA, /*neg_b=*/false, bB,
            /*c_mod=*/(short)0, acc, /*reuse_a=*/false, /*reuse_b=*/false);
    }

    // Per-lane sum over the 8 D VGPRs (rows). Column 0 totals then live in
    // lanes {0,16}, column 1 totals in lanes {1,17}.
    float s8 = acc[0] + acc[1] + acc[2] + acc[3] + acc[4] + acc[5] + acc[6] + acc[7];
    float l1w = __shfl(s8, 0, 32) + __shfl(s8, 16, 32);
    float l0w = __shfl(s8, 1, 32) + __shfl(s8, 17, 32);
    float tcw = tc;
#pragma unroll
    for (int o = 16; o > 0; o >>= 1) tcw += __shfl_xor(tcw, o, 32);

    __shared__ float lsum[WG / 32][3];
    if (lane == 0) {
        lsum[wave][0] = l1w;
        lsum[wave][1] = l0w;
        lsum[wave][2] = tcw;
    }
    __syncthreads();
    if (tid == 0) {
        float a = 0.f, b = 0.f, c = 0.f;
        for (int w = 0; w < WG / 32; ++w) {  // fixed order -> deterministic
            a += lsum[w][0];
            b += lsum[w][1];
            c += lsum[w][2];
        }
        ws[blockIdx.x * 3 + 0] = a;
        ws[blockIdx.x * 3 + 1] = b;
        ws[blockIdx.x * 3 + 2] = c;
    }
}

// Kernel 2: single block. Handles any scalar remainder (none for this shape),
// reduces the per-block partials in a fixed tree, applies the loss formula.
__global__ __launch_bounds__(WG) void bce_final_kernel(
    const float* __restrict__ x, const int* __restrict__ t,
    const float* __restrict__ ws, float* __restrict__ out,
    long long n, long long rem_start, int nblocks)
{
    const int tid = threadIdx.x;
    double r1 = 0.0, r0 = 0.0, rt = 0.0;

    for (long long i = rem_start + tid; i < n; i += WG) {
        float xe = x[i];
        int   te = t[i];
        float arg = te ? xe : (1.0f - xe);
        float lg  = __logf(arg);
        if (te) { r1 += (double)lg; rt += 1.0; }
        else    { r0 += (double)lg; }
    }
    for (int b = tid; b < nblocks; b += WG) {
        r1 += (double)ws[b * 3 + 0];
        r0 += (double)ws[b * 3 + 1];
        rt += (double)ws[b * 3 + 2];
    }

    __shared__ double s1[WG], s0[WG], st[WG];
    s1[tid] = r1; s0[tid] = r0; st[tid] = rt;
    __syncthreads();
    for (int off = WG / 2; off > 0; off >>= 1) {
        if (tid < off) {
            s1[tid] += s1[tid + off];
            s0[tid] += s0[tid + off];
            st[tid] += st[tid + off];
        }
        __syncthreads();
    }
    if (tid == 0) {
        double loss1 = s1[0], loss0 = s0[0], s = st[0];
        double len = (double)n + 1.0;
        double wp = len / (s + 1.0);
        double wn = len / (len - s);
        double loss = (-wp * loss1 - wn * loss0) / (double)n;
        out[0] = (float)loss;
    }
}

extern "C" void kernel_launch(void* const* d_in, const int* in_sizes, int n_in,
                              void* d_out, int out_size, void* d_ws, size_t ws_size,
                              hipStream_t stream) {
    const float* x = (const float*)d_in[0];
    const int*   t = (const int*)d_in[1];
    long long n  = (long long)in_sizes[0];
    long long n4 = n / 4;
    long long T  = (long long)GRID * WG;
    int cnt = (int)(n4 / T);                 // = 32 for B=32768, L=1000 (exact)
    long long rem_start = (long long)cnt * T * 4;  // = n for this shape
    float* ws = (float*)d_ws;                // 1000 blocks * 3 floats = 12 KB

    bce_partial_kernel<<<GRID, WG, 0, stream>>>(x, t, ws, cnt);
    bce_final_kernel<<<1, WG, 0, stream>>>(x, t, ws, (float*)d_out,
                                           n, rem_start, GRID);
}